// LoRAModulatedAllegroLayer_10720238371312
// MI455X (gfx1250) — compile-verified
//
#include <hip/hip_runtime.h>
#include <hip/hip_bf16.h>
#include <math.h>

// ---- problem constants --------------------------------------------------
#define E_EDGES   262144
#define N_NODES   8192
#define TILES     (E_EDGES / 16)

typedef float v2f __attribute__((ext_vector_type(2)));
typedef float v8f __attribute__((ext_vector_type(8)));

__device__ __forceinline__ v8f wmma_f32(v2f a, v2f b, v8f c) {
  // D = A(16x4) * B(4x16) + C(16x16), all f32, wave32
  return __builtin_amdgcn_wmma_f32_16x16x4_f32(
      /*neg_a=*/false, a, /*neg_b=*/false, b,
      /*c_mod=*/(short)0, c, /*reuse_a=*/false, /*reuse_b=*/false);
}

__device__ __forceinline__ float silu(float v) {
  return v / (1.0f + __expf(-v));
}

// ---- kernel 0: zero node accumulators ----------------------------------
__global__ void zero_kernel(float* __restrict__ p, long n) {
  long i = (long)blockIdx.x * blockDim.x + threadIdx.x;
  long stride = (long)gridDim.x * blockDim.x;
  for (; i < n; i += stride) p[i] = 0.0f;
}

// ---- kernel 1: s_l = mod @ M_l  (4 layers x rank 4) --------------------
__global__ void seff_kernel(const float* __restrict__ mod,
                            const float* __restrict__ M0,
                            const float* __restrict__ M1,
                            const float* __restrict__ M2,
                            const float* __restrict__ M3,
                            float* __restrict__ s) {
  int t = threadIdx.x;
  if (t >= 16) return;
  const float* M = (t < 4) ? M0 : (t < 8) ? M1 : (t < 12) ? M2 : M3;
  int r = t & 3;
  float acc = 0.0f;
  for (int k = 0; k < 128; ++k) acc += mod[k] * M[k * 4 + r];
  s[t] = acc;
}

// ---- kernel 2: Weff = (W + A diag(s) B) / sqrt(din) --------------------
__global__ void weff_kernel(const float* __restrict__ W,
                            const float* __restrict__ A,
                            const float* __restrict__ B,
                            const float* __restrict__ s,
                            float* __restrict__ out,
                            int din, int dout, float scale) {
  int idx = blockIdx.x * blockDim.x + threadIdx.x;
  int total = din * dout;
  if (idx >= total) return;
  int i = idx / dout;
  int j = idx - i * dout;
  float d = 0.0f;
#pragma unroll
  for (int r = 0; r < 4; ++r) d += A[i * 4 + r] * s[r] * B[r * dout + j];
  out[idx] = (W[idx] + d) * scale;
}

// ---- kernel 3: message pass — w = xm @ W1eff, scatter w*Y*m ------------
// 128 threads = 4 waves; wave w owns output column tile w (cols 16w..16w+15)
__global__ __launch_bounds__(128) void msg_kernel(
    const float* __restrict__ x, const float* __restrict__ vectors,
    const float* __restrict__ m, const int* __restrict__ senders,
    const float* __restrict__ W1eff /*128x64*/,
    float* __restrict__ nodeWY /*N*64*4*/, float* __restrict__ nodeCnt) {
  __shared__ float sXM[16 * 132];
  __shared__ float sY[16][4];
  __shared__ float sM[16];
  __shared__ int   sSend[16];

  const int tid  = threadIdx.x;
  const int wv   = tid >> 5;         // wave id 0..3
  const int lane = tid & 31;
  const int h = lane >> 4, l16 = lane & 15;
  const long e0 = (long)blockIdx.x * 16;

  if (tid < 16) {
    long e = e0 + tid;
    float vx = vectors[e * 3 + 0], vy = vectors[e * 3 + 1], vz = vectors[e * 3 + 2];
    float r = sqrtf(vx * vx + vy * vy + vz * vz);
    float inv = 1.7320508075688772f / fmaxf(r, 1e-12f); // sqrt(3)/max(r,eps)
    sY[tid][0] = 1.0f;
    sY[tid][1] = vx * inv;
    sY[tid][2] = vy * inv;
    sY[tid][3] = vz * inv;
    float mm = m[e];
    sM[tid] = mm;
    int sd = senders[e];
    sSend[tid] = sd;
    atomicAdd(&nodeCnt[sd], mm); // n_nb segment sum
  }
  __syncthreads();

  for (int i = tid; i < 16 * 128; i += 128) {
    int e = i >> 7, c = i & 127;
    sXM[e * 132 + c] = x[(e0 + e) * 128 + c] * sM[e];
  }
  __syncthreads();

  // GEMM: (16 x 128) @ (128 x 64); this wave computes cols [16*wv, 16*wv+16)
  v8f acc = {};
  for (int k0 = 0; k0 < 128; k0 += 4) {
    int ka = k0 + 2 * h;
    v2f a;
    a.x = sXM[l16 * 132 + ka];
    a.y = sXM[l16 * 132 + ka + 1];
    v2f b;
    b.x = W1eff[ka * 64 + wv * 16 + l16];
    b.y = W1eff[(ka + 1) * 64 + wv * 16 + l16];
    acc = wmma_f32(a, b, acc);
  }

  // scatter msg = w * Y * m into nodeWY via f32 atomics (L2-resident)
#pragma unroll
  for (int r = 0; r < 8; ++r) {
    int e = r + 8 * h;
    int c = wv * 16 + l16;
    float wvv = acc[r] * sM[e];
    float* base = nodeWY + (size_t)sSend[e] * 256 + c * 4;
    atomicAdd(base + 0, wvv * sY[e][0]);
    atomicAdd(base + 1, wvv * sY[e][1]);
    atomicAdd(base + 2, wvv * sY[e][2]);
    atomicAdd(base + 3, wvv * sY[e][3]);
  }
}

// ---- kernel 4: gather, tensor products, MLP, outputs -------------------
// 128 threads = 4 waves; wave w owns col tiles {2w,2w+1} of 128-wide layers
// and col tile w of the 64-wide V_out GEMM.
__global__ __launch_bounds__(128) void fused_kernel(
    const float* __restrict__ x, const float* __restrict__ V,
    const float* __restrict__ u, const float* __restrict__ m,
    const int* __restrict__ senders,
    const float* __restrict__ nodeWY, const float* __restrict__ nodeCnt,
    const float* __restrict__ L0 /*256x128*/, const float* __restrict__ L1 /*128x128*/,
    const float* __restrict__ L2 /*128x128*/, const float* __restrict__ linW /*192x64*/,
    float* __restrict__ xout, float* __restrict__ vout) {
  __shared__ float sFeat[16 * 260];  // x2 (256 wide), later reused for V1 component
  __shared__ float sWY[16 * 260];    // gathered wY, 64*4
  __shared__ float sVC[16 * 260];    // V reshaped, 64*4
  __shared__ float sH[16 * 132];     // MLP hidden
  __shared__ int   sSend[16];
  __shared__ float sInv[16];
  __shared__ float sU[16];
  __shared__ float sMm[16];

  const int tid  = threadIdx.x;
  const int wv   = tid >> 5;
  const int lane = tid & 31;
  const int h = lane >> 4, l16 = lane & 15;
  const long e0 = (long)blockIdx.x * 16;

  if (tid < 16) {
    long e = e0 + tid;
    int sd = senders[e];
    sSend[tid] = sd;
    sInv[tid] = 1.0f / (nodeCnt[sd] + 1e-5f);
    sU[tid] = u[e];
    sMm[tid] = m[e];
  }
  __syncthreads();

  for (int i = tid; i < 16 * 128; i += 128) {
    int e = i >> 7, c = i & 127;
    sFeat[e * 260 + c] = x[(e0 + e) * 128 + c] * sMm[e];
  }
  for (int i = tid; i < 16 * 256; i += 128) {
    int e = i >> 8, c = i & 255;
    sWY[e * 260 + c] = nodeWY[(size_t)sSend[e] * 256 + c] * sInv[e];
    sVC[e * 260 + c] = V[(e0 + e) * 256 + c];
  }
  __syncthreads();

  // scalar tensor products o0a, o0b -> sFeat[:,128:256]
  const float inv_s3 = 0.5773502691896258f; // 1/sqrt(3)
  for (int i = tid; i < 16 * 64; i += 128) {
    int e = i >> 6, c = i & 63;
    const float* wy = &sWY[e * 260 + c * 4];
    const float* vc = &sVC[e * 260 + c * 4];
    sFeat[e * 260 + 128 + c] = wy[0] * vc[0];
    sFeat[e * 260 + 192 + c] =
        (wy[1] * vc[1] + wy[2] * vc[2] + wy[3] * vc[3]) * inv_s3;
  }
  __syncthreads();

  // ---- MLP layer 0: (16x256) @ (256x128), SiLU ----
  v8f acc[2] = {};
  for (int k0 = 0; k0 < 256; k0 += 4) {
    int ka = k0 + 2 * h;
    v2f a;
    a.x = sFeat[l16 * 260 + ka];
    a.y = sFeat[l16 * 260 + ka + 1];
#pragma unroll
    for (int j = 0; j < 2; ++j) {
      int t = 2 * wv + j;
      v2f b;
      b.x = L0[ka * 128 + t * 16 + l16];
      b.y = L0[(ka + 1) * 128 + t * 16 + l16];
      acc[j] = wmma_f32(a, b, acc[j]);
    }
  }
#pragma unroll
  for (int j = 0; j < 2; ++j)
#pragma unroll
    for (int r = 0; r < 8; ++r)
      sH[(r + 8 * h) * 132 + (2 * wv + j) * 16 + l16] = silu(acc[j][r]);
  __syncthreads();

  // ---- MLP layer 1: (16x128) @ (128x128), SiLU ----
#pragma unroll
  for (int j = 0; j < 2; ++j) acc[j] = v8f{};
  for (int k0 = 0; k0 < 128; k0 += 4) {
    int ka = k0 + 2 * h;
    v2f a;
    a.x = sH[l16 * 132 + ka];
    a.y = sH[l16 * 132 + ka + 1];
#pragma unroll
    for (int j = 0; j < 2; ++j) {
      int t = 2 * wv + j;
      v2f b;
      b.x = L1[ka * 128 + t * 16 + l16];
      b.y = L1[(ka + 1) * 128 + t * 16 + l16];
      acc[j] = wmma_f32(a, b, acc[j]);
    }
  }
  __syncthreads(); // all reads of sH complete before overwrite
#pragma unroll
  for (int j = 0; j < 2; ++j)
#pragma unroll
    for (int r = 0; r < 8; ++r)
      sH[(r + 8 * h) * 132 + (2 * wv + j) * 16 + l16] = silu(acc[j][r]);
  __syncthreads();

  // ---- MLP layer 2: (16x128) @ (128x128), then x_out = u*h ----
#pragma unroll
  for (int j = 0; j < 2; ++j) acc[j] = v8f{};
  for (int k0 = 0; k0 < 128; k0 += 4) {
    int ka = k0 + 2 * h;
    v2f a;
    a.x = sH[l16 * 132 + ka];
    a.y = sH[l16 * 132 + ka + 1];
#pragma unroll
    for (int j = 0; j < 2; ++j) {
      int t = 2 * wv + j;
      v2f b;
      b.x = L2[ka * 128 + t * 16 + l16];
      b.y = L2[(ka + 1) * 128 + t * 16 + l16];
      acc[j] = wmma_f32(a, b, acc[j]);
    }
  }
#pragma unroll
  for (int j = 0; j < 2; ++j)
#pragma unroll
    for (int r = 0; r < 8; ++r) {
      int e = r + 8 * h;
      xout[(e0 + e) * 128 + (2 * wv + j) * 16 + l16] = sU[e] * acc[j][r];
    }

  // ---- V_out: for each component i, build V1_i (16x192) and GEMM linW ----
  const float inv_s2 = 0.7071067811865475f;    // 1/sqrt(2)
  const float inv_s192 = 0.07216878364870323f; // 1/sqrt(192)
  for (int comp = 0; comp < 3; ++comp) {
    __syncthreads(); // previous reads of sFeat complete before overwrite
    for (int e = 0; e < 16; ++e) {
      const float* wy = &sWY[e * 260];
      const float* vc = &sVC[e * 260];
      for (int c = tid; c < 192; c += 128) {
        float val;
        if (c < 64) {                       // o1a = s1 * v2
          val = wy[4 * c] * vc[4 * c + 1 + comp];
        } else if (c < 128) {               // o1b = v1 * s2
          int cc = c - 64;
          val = wy[4 * cc + 1 + comp] * vc[4 * cc];
        } else {                            // o1c = cross(v1,v2)/sqrt(2)
          int cc = c - 128;
          int i1 = (comp == 2) ? 0 : comp + 1;
          int i2 = (comp == 0) ? 2 : comp - 1;
          val = (wy[4 * cc + 1 + i1] * vc[4 * cc + 1 + i2] -
                 wy[4 * cc + 1 + i2] * vc[4 * cc + 1 + i1]) * inv_s2;
        }
        sFeat[e * 260 + c] = val;
      }
    }
    __syncthreads();

    // (16x192) @ (192x64); this wave computes cols [16*wv, 16*wv+16)
    v8f vacc = {};
    for (int k0 = 0; k0 < 192; k0 += 4) {
      int ka = k0 + 2 * h;
      v2f a;
      a.x = sFeat[l16 * 260 + ka];
      a.y = sFeat[l16 * 260 + ka + 1];
      v2f b;
      b.x = linW[ka * 64 + wv * 16 + l16];
      b.y = linW[(ka + 1) * 64 + wv * 16 + l16];
      vacc = wmma_f32(a, b, vacc);
    }
#pragma unroll
    for (int r = 0; r < 8; ++r) {
      int e = r + 8 * h;
      int o = wv * 16 + l16;
      vout[((e0 + e) * 64 + o) * 3 + comp] = vacc[r] * inv_s192;
    }
  }
}

// ---- host launch --------------------------------------------------------
extern "C" void kernel_launch(void* const* d_in, const int* in_sizes, int n_in,
                              void* d_out, int out_size, void* d_ws, size_t ws_size,
                              hipStream_t stream) {
  (void)in_sizes; (void)n_in; (void)out_size; (void)ws_size;

  const float* vectors = (const float*)d_in[0];
  const float* x       = (const float*)d_in[1];
  const float* V       = (const float*)d_in[2];
  const float* u       = (const float*)d_in[3];
  const float* m       = (const float*)d_in[4];
  const int*   senders = (const int*)d_in[5];
  const float* mod     = (const float*)d_in[6];
  const float* W1 = (const float*)d_in[7];
  const float* A1 = (const float*)d_in[8];
  const float* B1 = (const float*)d_in[9];
  const float* M1 = (const float*)d_in[10];
  const float* Wl0 = (const float*)d_in[11];
  const float* Al0 = (const float*)d_in[12];
  const float* Bl0 = (const float*)d_in[13];
  const float* Ml0 = (const float*)d_in[14];
  const float* Wl1 = (const float*)d_in[15];
  const float* Al1 = (const float*)d_in[16];
  const float* Bl1 = (const float*)d_in[17];
  const float* Ml1 = (const float*)d_in[18];
  const float* Wl2 = (const float*)d_in[19];
  const float* Al2 = (const float*)d_in[20];
  const float* Bl2 = (const float*)d_in[21];
  const float* Ml2 = (const float*)d_in[22];
  const float* linW = (const float*)d_in[23];

  // workspace layout (floats)
  float* ws      = (float*)d_ws;
  float* s4      = ws;                 // 16
  float* W1eff   = ws + 16;            // 128*64
  float* L0eff   = W1eff + 128 * 64;   // 256*128
  float* L1eff   = L0eff + 256 * 128;  // 128*128
  float* L2eff   = L1eff + 128 * 128;  // 128*128
  float* nodeCnt = L2eff + 128 * 128;  // N
  float* nodeWY  = nodeCnt + N_NODES;  // N*256

  const float rs128 = 0.08838834764831845f; // 1/sqrt(128)
  const float rs256 = 0.0625f;              // 1/sqrt(256)

  zero_kernel<<<2048, 256, 0, stream>>>(nodeCnt, (long)N_NODES + (long)N_NODES * 256);
  seff_kernel<<<1, 16, 0, stream>>>(mod, M1, Ml0, Ml1, Ml2, s4);
  weff_kernel<<<(128 * 64 + 255) / 256, 256, 0, stream>>>(W1, A1, B1, s4 + 0, W1eff, 128, 64, rs128);
  weff_kernel<<<(256 * 128 + 255) / 256, 256, 0, stream>>>(Wl0, Al0, Bl0, s4 + 4, L0eff, 256, 128, rs256);
  weff_kernel<<<(128 * 128 + 255) / 256, 256, 0, stream>>>(Wl1, Al1, Bl1, s4 + 8, L1eff, 128, 128, rs128);
  weff_kernel<<<(128 * 128 + 255) / 256, 256, 0, stream>>>(Wl2, Al2, Bl2, s4 + 12, L2eff, 128, 128, rs128);

  msg_kernel<<<TILES, 128, 0, stream>>>(x, vectors, m, senders, W1eff, nodeWY, nodeCnt);

  float* xout = (float*)d_out;
  float* vout = xout + (size_t)E_EDGES * 128;
  fused_kernel<<<TILES, 128, 0, stream>>>(x, V, u, m, senders, nodeWY, nodeCnt,
                                          L0eff, L1eff, L2eff, linW, xout, vout);
}